// HeteroGraphSAGE_45088566673663
// MI455X (gfx1250) — compile-verified
//
#include <hip/hip_runtime.h>
#include <hip/hip_bf16.h>

typedef __attribute__((ext_vector_type(2))) float v2f;
typedef __attribute__((ext_vector_type(8))) float v8f;

#define C_DIM 128

// ---------------------------------------------------------------------------
// Zero a float4-aligned region (segment sums + counts) each call.
// ---------------------------------------------------------------------------
__global__ __launch_bounds__(256) void zero_kernel(float4* __restrict__ p, long n4) {
    long i = (long)blockIdx.x * blockDim.x + threadIdx.x;
    if (i < n4) p[i] = make_float4(0.f, 0.f, 0.f, 0.f);
}

// ---------------------------------------------------------------------------
// Edge scatter: one wave (32 lanes) per edge; each lane handles 4 features.
// s[dst,:] += x_src[src,:]; cnt[dst] += 1
// ---------------------------------------------------------------------------
__global__ __launch_bounds__(256) void scatter_add_kernel(
    const float* __restrict__ xsrc, const int* __restrict__ ei,
    float* __restrict__ s, float* __restrict__ cnt, int E)
{
    const int wave = threadIdx.x >> 5;
    const int lane = threadIdx.x & 31;
    const int e = blockIdx.x * 8 + wave;
    if (e >= E) return;
    const int src = ei[e];        // ei[0, e]
    const int dst = ei[E + e];    // ei[1, e]
    const float4 v = *(const float4*)(xsrc + (size_t)src * C_DIM + lane * 4);
    float* p = s + (size_t)dst * C_DIM + lane * 4;
    atomicAdd(p + 0, v.x);
    atomicAdd(p + 1, v.y);
    atomicAdd(p + 2, v.z);
    atomicAdd(p + 3, v.w);
    if (lane == 0) atomicAdd(cnt + dst, 1.0f);
}

// ---------------------------------------------------------------------------
// out = (s/max(cnt,1)) @ Wl^T + bias_l + xdst @ Wr^T       (fp32 WMMA)
// Block: 256 threads = 8 waves. Wave tile: 16 rows x 64 cols (4 x 16x16).
// Block tile: 128 rows x 64 cols; grid = (ceil(N/128), 2).
// LDS holds transposed weight strips in K-pair-interleaved layout:
//   lw[(k2*64 + n)*2 + j] = W[(n0+n)*128 + 2*k2 + j]
// so each B fragment is ONE contiguous ds_load_b64 -> even VGPR pair.
// ---------------------------------------------------------------------------
__global__ __launch_bounds__(256) void sage_gemm_kernel(
    const float* __restrict__ s,       // N x 128 segment sums
    const float* __restrict__ cnt,     // N counts (float)
    const float* __restrict__ xdst,    // N x 128 self features
    const float* __restrict__ Wl,      // 128 x 128 row-major (out, in)
    const float* __restrict__ bias_l,  // 128
    const float* __restrict__ Wr,      // 128 x 128 row-major
    float* __restrict__ out,           // N x 128
    int n_rows)
{
    extern __shared__ float smem[];
    float* lwl = smem;           // 64 k-pairs x 64 cols x 2 = 8192 floats
    float* lwr = smem + 8192;

    const int tid = threadIdx.x;
    const int n0 = blockIdx.y * 64;

    // Stage weight strips: n-fast indexing -> conflict-free ds_store_b64
    // (global reads are strided but the 64KB weight matrices are L2-hot).
    for (int idx = tid; idx < 4096; idx += 256) {
        const int n  = idx & 63;
        const int k2 = idx >> 6;
        *(v2f*)&lwl[(k2 * 64 + n) * 2] = *(const v2f*)&Wl[(n0 + n) * C_DIM + k2 * 2];
        *(v2f*)&lwr[(k2 * 64 + n) * 2] = *(const v2f*)&Wr[(n0 + n) * C_DIM + k2 * 2];
    }
    __syncthreads();

    const int wave = tid >> 5;
    const int lane = tid & 31;
    const int hh   = lane >> 4;    // half-wave select: K offset 0 or 2
    const int l16  = lane & 15;
    const int m0   = blockIdx.x * 128 + wave * 16;

    // A-fragment row for this lane (same row both halves; half picks K pair)
    int arow = m0 + l16;
    if (arow > n_rows - 1) arow = n_rows - 1;       // clamp tail loads
    const float invc = 1.0f / fmaxf(cnt[arow], 1.0f);
    const float* __restrict__ srow = s    + (size_t)arow * C_DIM;
    const float* __restrict__ xrow = xdst + (size_t)arow * C_DIM;
    __builtin_prefetch(srow, 0, 0);
    __builtin_prefetch(xrow, 0, 0);

    v8f acc[4];
#pragma unroll
    for (int t = 0; t < 4; ++t) acc[t] = (v8f){0.f,0.f,0.f,0.f,0.f,0.f,0.f,0.f};

#pragma unroll 2
    for (int k0 = 0; k0 < C_DIM; k0 += 4) {
        const int kp = (k0 >> 1) + hh;              // this half-wave's k-pair idx
        // A fragments: 16x4 fp32 (lanes 0-15: K=k0,k0+1; lanes 16-31: K=k0+2,k0+3)
        v2f a_mean = *(const v2f*)(srow + k0 + hh * 2);
        a_mean.x *= invc; a_mean.y *= invc;
        const v2f a_self = *(const v2f*)(xrow + k0 + hh * 2);

        const float* blp = lwl + (size_t)kp * 128;  // 64 cols * 2
        const float* brp = lwr + (size_t)kp * 128;
#pragma unroll
        for (int t = 0; t < 4; ++t) {
            const int nc = t * 16 + l16;
            const v2f b_l = *(const v2f*)(blp + nc * 2);   // one ds_load_b64
            acc[t] = __builtin_amdgcn_wmma_f32_16x16x4_f32(
                false, a_mean, false, b_l, (short)0, acc[t], false, false);
            const v2f b_r = *(const v2f*)(brp + nc * 2);
            acc[t] = __builtin_amdgcn_wmma_f32_16x16x4_f32(
                false, a_self, false, b_r, (short)0, acc[t], false, false);
        }
    }

    // D layout: VGPR i -> row m0+i (lanes 0-15) / m0+i+8 (lanes 16-31), col n0+l16
    const bool full_tile = (m0 + 16) <= n_rows;     // wave-uniform
    if (full_tile) {
#pragma unroll
        for (int t = 0; t < 4; ++t) {
            const int col = n0 + t * 16 + l16;
            const float bias = bias_l[col];
#pragma unroll
            for (int i = 0; i < 8; ++i) {
                const int row = m0 + i + hh * 8;
                out[(size_t)row * C_DIM + col] = acc[t][i] + bias;
            }
        }
    } else {
#pragma unroll
        for (int t = 0; t < 4; ++t) {
            const int col = n0 + t * 16 + l16;
            const float bias = bias_l[col];
#pragma unroll
            for (int i = 0; i < 8; ++i) {
                const int row = m0 + i + hh * 8;
                if (row < n_rows) out[(size_t)row * C_DIM + col] = acc[t][i] + bias;
            }
        }
    }
}

// ---------------------------------------------------------------------------
// y = relu( (x - mu) * rsqrt(var + eps) * g + b ), per-row over 128 features.
// One wave per row; float4 per lane; wave32 shuffle reduction.
// ---------------------------------------------------------------------------
__global__ __launch_bounds__(256) void ln_relu_kernel(
    const float* __restrict__ x, const float* __restrict__ g,
    const float* __restrict__ b, float* __restrict__ y, int n_rows)
{
    const int wave = threadIdx.x >> 5;
    const int lane = threadIdx.x & 31;
    const int row = blockIdx.x * 8 + wave;
    if (row >= n_rows) return;

    const float4 v = *(const float4*)(x + (size_t)row * C_DIM + lane * 4);
    float s1 = v.x + v.y + v.z + v.w;
    float s2 = v.x * v.x + v.y * v.y + v.z * v.z + v.w * v.w;
#pragma unroll
    for (int off = 16; off > 0; off >>= 1) {
        s1 += __shfl_xor(s1, off, 32);
        s2 += __shfl_xor(s2, off, 32);
    }
    const float mu  = s1 * (1.0f / 128.0f);
    const float var = s2 * (1.0f / 128.0f) - mu * mu;
    const float r   = rsqrtf(var + 1e-5f);

    const float4 gg = *(const float4*)(g + lane * 4);
    const float4 bb = *(const float4*)(b + lane * 4);
    float4 o;
    o.x = fmaxf((v.x - mu) * r * gg.x + bb.x, 0.0f);
    o.y = fmaxf((v.y - mu) * r * gg.y + bb.y, 0.0f);
    o.z = fmaxf((v.z - mu) * r * gg.z + bb.z, 0.0f);
    o.w = fmaxf((v.w - mu) * r * gg.w + bb.w, 0.0f);
    *(float4*)(y + (size_t)row * C_DIM + lane * 4) = o;
}

// ---------------------------------------------------------------------------
extern "C" void kernel_launch(void* const* d_in, const int* in_sizes, int n_in,
                              void* d_out, int out_size, void* d_ws, size_t ws_size,
                              hipStream_t stream) {
    const int N = in_sizes[0] / C_DIM;
    const int E = in_sizes[2] / 2;
    const size_t NC = (size_t)N * C_DIM;

    const float* x_user0 = (const float*)d_in[0];
    const float* x_item0 = (const float*)d_in[1];
    const int*   ei_ui   = (const int*)d_in[2];
    const int*   ei_iu   = (const int*)d_in[3];
    float* out_f = (float*)d_out;

    // Workspace layout (floats)
    float* ws       = (float*)d_ws;
    float* s_user   = ws;                 // NC
    float* s_item   = ws + NC;            // NC
    float* cnt_user = ws + 2 * NC;        // N
    float* cnt_item = cnt_user + N;       // N
    float* out_user = cnt_item + N;       // NC
    float* out_item = out_user + NC;      // NC
    float* xu_buf   = out_item + NC;      // NC
    float* xi_buf   = xu_buf + NC;        // NC

    const float* xu_cur = x_user0;
    const float* xi_cur = x_item0;

    const long  zero_n4 = (long)(2 * NC + 2 * (size_t)N) / 4;
    const dim3  zgrid((unsigned)((zero_n4 + 255) / 256));
    const dim3  sgrid((unsigned)((E + 7) / 8));
    const dim3  ggrid((unsigned)((N + 127) / 128), 2);
    const dim3  lgrid((unsigned)((N + 7) / 8));
    const size_t lds_bytes = 2 * 8192 * sizeof(float);   // 64 KB dynamic LDS

    for (int l = 0; l < 2; ++l) {
        const float* Wl_ui = (const float*)d_in[4 + 10 * l + 0];
        const float* bl_ui = (const float*)d_in[4 + 10 * l + 1];
        const float* Wr_ui = (const float*)d_in[4 + 10 * l + 2];
        const float* Wl_iu = (const float*)d_in[4 + 10 * l + 3];
        const float* bl_iu = (const float*)d_in[4 + 10 * l + 4];
        const float* Wr_iu = (const float*)d_in[4 + 10 * l + 5];
        const float* g_u   = (const float*)d_in[4 + 10 * l + 6];
        const float* b_u   = (const float*)d_in[4 + 10 * l + 7];
        const float* g_i   = (const float*)d_in[4 + 10 * l + 8];
        const float* b_i   = (const float*)d_in[4 + 10 * l + 9];

        float* yu = (l == 0) ? xu_buf : out_f;
        float* yi = (l == 0) ? xi_buf : out_f + NC;

        zero_kernel<<<zgrid, 256, 0, stream>>>((float4*)s_user, zero_n4);

        // user -> item aggregation, item -> user aggregation
        scatter_add_kernel<<<sgrid, 256, 0, stream>>>(xu_cur, ei_ui, s_item, cnt_item, E);
        scatter_add_kernel<<<sgrid, 256, 0, stream>>>(xi_cur, ei_iu, s_user, cnt_user, E);

        sage_gemm_kernel<<<ggrid, 256, lds_bytes, stream>>>(
            s_item, cnt_item, xi_cur, Wl_ui, bl_ui, Wr_ui, out_item, N);
        sage_gemm_kernel<<<ggrid, 256, lds_bytes, stream>>>(
            s_user, cnt_user, xu_cur, Wl_iu, bl_iu, Wr_iu, out_user, N);

        ln_relu_kernel<<<lgrid, 256, 0, stream>>>(out_user, g_u, b_u, yu, N);
        ln_relu_kernel<<<lgrid, 256, 0, stream>>>(out_item, g_i, b_i, yi, N);

        xu_cur = yu;
        xi_cur = yi;
    }
}